// ILRMA_NOISY_37546604101778
// MI455X (gfx1250) — compile-verified
//
#include <hip/hip_runtime.h>

// ---------------------------------------------------------------------------
// ILRMA-noisy for MI455X (gfx1250, wave32).
// Planes r_s, r_n, p1, pn kept in d_ws; Lambda recomputed on the fly
// (fma instead of a 16.8MB plane round-trip; working set ~202MB ~ L2-resident).
// V-updates use V_WMMA_F32_16X16X4_F32 (contraction over I with K=8 outputs),
// with the T factor staged zero-padded in LDS so the A-fragment read is a
// single unconditional ds_load_b32 (no exec-mask divergence).
// All hot-loop reciprocals/sqrts use hardware v_rcp_f32 / v_sqrt_f32.
// Workspace requirement: 12*I*J floats + ~100K floats  (~202 MB).
// ---------------------------------------------------------------------------

#define I_N 1025
#define J_N 4096
#define K_N 8
#define N_ITER 8
#define SWITCH_ITER 5
#define ALPHA 0.1f
#define BETA 0.0f
#define EPS 1e-10f

typedef float v2f __attribute__((ext_vector_type(2)));
typedef float v8f __attribute__((ext_vector_type(8)));

static __device__ __forceinline__ float frcp(float x) { return __builtin_amdgcn_rcpf(x); }
static __device__ __forceinline__ float fsqrt(float x) { return __builtin_amdgcn_sqrtf(x); }
static __device__ __forceinline__ float frsq(float x) { return __builtin_amdgcn_rsqf(x); }

static __device__ __forceinline__ float lam_of(float rs, float rn, float l) {
  return fmaxf(fmaf(l, rn, rs), EPS);
}
static __device__ __forceinline__ float2 cmulf(float2 a, float2 b) {
  return make_float2(a.x * b.x - a.y * b.y, a.x * b.y + a.y * b.x);
}
static __device__ __forceinline__ float2 conjf2(float2 a) { return make_float2(a.x, -a.y); }
static __device__ __forceinline__ float2 caddf(float2 a, float2 b) { return make_float2(a.x + b.x, a.y + b.y); }
static __device__ __forceinline__ float2 cscalef(float2 a, float s) { return make_float2(a.x * s, a.y * s); }
// conj(a)*x
static __device__ __forceinline__ float2 conjmul(float2 a, float2 x) {
  return make_float2(a.x * x.x + a.y * x.y, a.x * x.y - a.y * x.x);
}

// ---- K0: transpose X (M,J,I,2) -> Xc,Y (I,M,J) float2, LDS-tiled ----------
__global__ __launch_bounds__(256) void k_transpose_init(const float2* __restrict__ X,
                                                        float2* __restrict__ Xc,
                                                        float2* __restrict__ Y) {
  __shared__ float2 tile[16][17];
  int tx = threadIdx.x, ty = threadIdx.y;
  int j0 = blockIdx.x * 16;
  int i0 = blockIdx.y * 16;
  int m = blockIdx.z;
  int i = i0 + tx;
  float2 v = make_float2(0.f, 0.f);
  if (i < I_N) v = X[((size_t)m * J_N + (j0 + ty)) * I_N + i];
  tile[ty][tx] = v;
  __syncthreads();
  int iw = i0 + ty;
  if (iw < I_N) {
    size_t o = ((size_t)iw * 2 + m) * J_N + (j0 + tx);
    float2 w = tile[tx][ty];
    Xc[o] = w;
    Y[o] = w;  // W = eye initially -> Y = Xc
  }
}

// ---- K0b: copy small factor matrices, lam_n = 1 ---------------------------
__global__ __launch_bounds__(256) void k_small_init(float* Ts, float* Tn, float* Vs, float* Vn,
                                                    float* lam,
                                                    const float* Ts_in, const float* Vs_in,
                                                    const float* Tn_in, const float* Vn_in) {
  int t = blockIdx.x * 256 + threadIdx.x;  // 128 blocks -> 32768 threads
  if (t < I_N * K_N) { Ts[t] = Ts_in[t]; Tn[t] = Tn_in[t]; }
  if (t < K_N * J_N) { Vs[t] = Vs_in[t]; Vn[t] = Vn_in[t]; }
  if (t < I_N) lam[t] = 1.0f;
}

// ---- K2: p1 = |Y0|^2, pn = |Y1|^2 -----------------------------------------
__global__ __launch_bounds__(256) void k_power(const float2* __restrict__ Y,
                                               float* __restrict__ p1, float* __restrict__ pn) {
  size_t t = (size_t)blockIdx.x * 256 + threadIdx.x;
  if (t >= (size_t)I_N * J_N) return;
  int i = (int)(t / J_N), j = (int)(t % J_N);
  float2 y0 = Y[((size_t)i * 2) * J_N + j];
  float2 y1 = Y[((size_t)i * 2 + 1) * J_N + j];
  p1[t] = y0.x * y0.x + y0.y * y0.y;
  pn[t] = y1.x * y1.x + y1.y * y1.y;
}

// ---- K5: r_s = max(Ts@Vs,eps), r_n = max(Tn@Vn,eps) (flag-selected) -------
__global__ __launch_bounds__(256) void k_recompute_r(int do_rs, int do_rn,
                                                     const float* __restrict__ Ts,
                                                     const float* __restrict__ Vs,
                                                     const float* __restrict__ Tn,
                                                     const float* __restrict__ Vn,
                                                     float* __restrict__ rs,
                                                     float* __restrict__ rn) {
  size_t t = (size_t)blockIdx.x * 256 + threadIdx.x;
  if (t >= (size_t)I_N * J_N) return;
  int i = (int)(t / J_N), j = (int)(t % J_N);
  if (do_rs) {
    float a = 0.f;
#pragma unroll
    for (int k = 0; k < K_N; ++k) a = fmaf(Ts[i * K_N + k], Vs[k * J_N + j], a);
    rs[t] = fmaxf(a, EPS);
  }
  if (do_rn) {
    float a = 0.f;
#pragma unroll
    for (int k = 0; k < K_N; ++k) a = fmaf(Tn[i * K_N + k], Vn[k * J_N + j], a);
    rn[t] = fmaxf(a, EPS);
  }
}

// ---- K4: row reductions over j. MODE 0: Ts update, 1: Tn update, 2: lam ---
template <int MODE>
__global__ __launch_bounds__(256) void k_rowreduce(float* __restrict__ T,
                                                   const float* __restrict__ V,
                                                   const float* __restrict__ rs,
                                                   const float* __restrict__ rn,
                                                   const float* __restrict__ p1,
                                                   const float* __restrict__ pn,
                                                   float* __restrict__ lam) {
  __shared__ float red[16 * 264];
  int i = blockIdx.x, tid = threadIdx.x;
  float lamv = lam[i];
  float num[8] = {};
  float den[8] = {};
  for (int j = tid; j < J_N; j += 256) {
    size_t idx = (size_t)i * J_N + j;
    float vrs = rs[idx], vrn = rn[idx], vp1 = p1[idx];
    float il = frcp(lam_of(vrs, vrn, lamv));
    float anum, aden;
    if (MODE == 0) {
      anum = vp1 * il * il;
      aden = il;
    } else if (MODE == 1) {
      float irn = frcp(vrn);
      float vpn = pn[idx];
      anum = fmaf(lamv * vp1, il * il, vpn * irn * irn);
      aden = fmaf(lamv, il, irn);  // (M-1) == 1
    } else {
      anum = vrn * vp1 * il * il;
      aden = vrn * il;
    }
    if (MODE == 2) {
      num[0] += anum;
      den[0] += aden;
    } else {
#pragma unroll
      for (int k = 0; k < K_N; ++k) {
        float v = V[k * J_N + j];
        num[k] = fmaf(anum, v, num[k]);
        den[k] = fmaf(aden, v, den[k]);
      }
    }
  }
#pragma unroll
  for (int k = 0; k < 8; ++k) {
    red[k * 264 + tid] = num[k];
    red[(k + 8) * 264 + tid] = den[k];
  }
  __syncthreads();
  for (int s = 128; s > 0; s >>= 1) {
    if (tid < s) {
#pragma unroll
      for (int q = 0; q < 16; ++q) red[q * 264 + tid] += red[q * 264 + tid + s];
    }
    __syncthreads();
  }
  if (tid == 0) {
    if (MODE == 2) {
      lam[i] = fmaxf(lamv * fsqrt(red[0] * frcp(red[8 * 264] + EPS)), EPS);
    } else {
#pragma unroll
      for (int k = 0; k < K_N; ++k) {
        float n = red[k * 264], d = red[(k + 8) * 264];
        T[i * K_N + k] = fmaxf(T[i * K_N + k] * fsqrt(n * frcp(d + EPS)), EPS);
      }
    }
  }
}

// ---- KW: V updates via V_WMMA_F32_16X16X4_F32 (contract over I) -----------
template <int MODE>
static __device__ __forceinline__ void bvals(size_t idx, float lamv,
                                             const float* __restrict__ rs,
                                             const float* __restrict__ rn,
                                             const float* __restrict__ p1,
                                             const float* __restrict__ pn,
                                             float& bn, float& bd) {
  float vrs = rs[idx], vrn = rn[idx], vp1 = p1[idx];
  float il = frcp(lam_of(vrs, vrn, lamv));
  if (MODE == 0) {
    bn = vp1 * il * il;
    bd = il;
  } else {
    float irn = frcp(vrn);
    float vpn = pn[idx];
    bn = fmaf(lamv * vp1, il * il, vpn * irn * irn);
    bd = fmaf(lamv, il, irn);
  }
}

template <int MODE>
__global__ __launch_bounds__(256) void k_vupdate_wmma(float* __restrict__ V,
                                                      const float* __restrict__ T,
                                                      const float* __restrict__ rs,
                                                      const float* __restrict__ rn,
                                                      const float* __restrict__ p1,
                                                      const float* __restrict__ pn,
                                                      const float* __restrict__ lam) {
  // T zero-padded to 16 columns in LDS: unconditional A-fragment loads.
  __shared__ float Tl[I_N * 16];  // 65.6 KB
  __shared__ float Ll[1056];      // lam_n
  for (int t = threadIdx.x; t < I_N * 16; t += 256) {
    int row = t >> 4, col = t & 15;
    float v = T[row * K_N + (col & (K_N - 1))];
    Tl[t] = (col < K_N) ? v : 0.f;
  }
  for (int t = threadIdx.x; t < I_N; t += 256) Ll[t] = lam[t];
  __syncthreads();

  int lane = threadIdx.x & 31;
  int wave = threadIdx.x >> 5;
  int j0 = (blockIdx.x * 8 + wave) * 16;  // 32 blocks x 8 waves x 16 j = 4096
  int mrow = lane & 15;                   // A row (= output k), B column (= j)
  int half = lane >> 4;
  int j = j0 + mrow;
  v8f cnum = {0.f, 0.f, 0.f, 0.f, 0.f, 0.f, 0.f, 0.f};
  v8f cden = {0.f, 0.f, 0.f, 0.f, 0.f, 0.f, 0.f, 0.f};
  for (int ic = 0; ic < 256; ++ic) {  // i = 0..1023 in chunks of 4
    int ia = ic * 4 + 2 * half;       // i for vgpr0; +1 for vgpr1
    v2f a;                            // A[m, icol] = T[i, m] (m<8), 0 pad
    a.x = Tl[ia * 16 + mrow];
    a.y = Tl[(ia + 1) * 16 + mrow];
    float b0n, b0d, b1n, b1d;
    bvals<MODE>((size_t)ia * J_N + j, Ll[ia], rs, rn, p1, pn, b0n, b0d);
    bvals<MODE>((size_t)(ia + 1) * J_N + j, Ll[ia + 1], rs, rn, p1, pn, b1n, b1d);
    v2f bn, bd;
    bn.x = b0n; bn.y = b1n;
    bd.x = b0d; bd.y = b1d;
    cnum = __builtin_amdgcn_wmma_f32_16x16x4_f32(false, a, false, bn, (short)0, cnum, false, false);
    cden = __builtin_amdgcn_wmma_f32_16x16x4_f32(false, a, false, bd, (short)0, cden, false, false);
  }
  if (lane < 16) {  // C layout: vgpr v holds k=v on lanes 0..15, col j0+lane
    int jr = j0 + lane;
    float brn, brd;  // remainder row i = 1024
    bvals<MODE>((size_t)1024 * J_N + jr, Ll[1024], rs, rn, p1, pn, brn, brd);
#pragma unroll
    for (int k = 0; k < K_N; ++k) {
      float tv = Tl[1024 * 16 + k];
      float n = cnum[k] + tv * brn;
      float d = cden[k] + tv * brd;
      size_t vi = (size_t)k * J_N + jr;
      V[vi] = fmaxf(V[vi] * fsqrt(n * frcp(d + EPS)), EPS);
    }
  }
}

// ---- K6: per-bin 2x2 weighted covariances G_s, G_n ------------------------
__global__ __launch_bounds__(256) void k_covariance(const float2* __restrict__ Xc,
                                                    const float* __restrict__ rs,
                                                    const float* __restrict__ rn,
                                                    const float* __restrict__ lam,
                                                    float* __restrict__ Gs,
                                                    float* __restrict__ Gn) {
  __shared__ float red[8 * 264];
  int i = blockIdx.x, tid = threadIdx.x;
  float lamv = lam[i];
  float acc[8] = {};
  for (int j = tid; j < J_N; j += 256) {
    size_t idx = (size_t)i * J_N + j;
    float2 x0 = Xc[((size_t)i * 2) * J_N + j];
    float2 x1 = Xc[((size_t)i * 2 + 1) * J_N + j];
    float ss = frcp(lam_of(rs[idx], rn[idx], lamv));
    float sn = frcp(rn[idx]);
    float a00 = x0.x * x0.x + x0.y * x0.y;
    float a11 = x1.x * x1.x + x1.y * x1.y;
    float re01 = x0.x * x1.x + x0.y * x1.y;
    float im01 = x0.y * x1.x - x0.x * x1.y;
    acc[0] += a00 * ss; acc[1] += a11 * ss; acc[2] += re01 * ss; acc[3] += im01 * ss;
    acc[4] += a00 * sn; acc[5] += a11 * sn; acc[6] += re01 * sn; acc[7] += im01 * sn;
  }
#pragma unroll
  for (int k = 0; k < 8; ++k) red[k * 264 + tid] = acc[k];
  __syncthreads();
  for (int s = 128; s > 0; s >>= 1) {
    if (tid < s) {
#pragma unroll
      for (int q = 0; q < 8; ++q) red[q * 264 + tid] += red[q * 264 + tid + s];
    }
    __syncthreads();
  }
  if (tid == 0) {
    const float sc = 1.f / (float)J_N;
    Gs[i * 4 + 0] = red[0 * 264] * sc + EPS;
    Gs[i * 4 + 1] = red[1 * 264] * sc + EPS;
    Gs[i * 4 + 2] = red[2 * 264] * sc;
    Gs[i * 4 + 3] = red[3 * 264] * sc;
    Gn[i * 4 + 0] = red[4 * 264] * sc + EPS;
    Gn[i * 4 + 1] = red[5 * 264] * sc + EPS;
    Gn[i * 4 + 2] = red[6 * 264] * sc;
    Gn[i * 4 + 3] = red[7 * 264] * sc;
  }
}

// ---- K7: closed-form 2x2 Cholesky -> Hermitian eig -> demix matrix W ------
__global__ __launch_bounds__(256) void k_eig2x2(const float* __restrict__ Gs,
                                                const float* __restrict__ Gn,
                                                float* __restrict__ Wm) {
  int i = blockIdx.x * 256 + threadIdx.x;
  if (i >= I_N) return;
  float a = Gs[i * 4 + 0], b = Gs[i * 4 + 1];
  float2 c = make_float2(Gs[i * 4 + 2], Gs[i * 4 + 3]);
  float n00 = Gn[i * 4 + 0], n11 = Gn[i * 4 + 1];
  float2 n01 = make_float2(Gn[i * 4 + 2], Gn[i * 4 + 3]);
  float asafe = fmaxf(a, 1e-20f);
  float i11 = frsq(asafe);
  float2 L10 = cscalef(conjf2(c), i11);
  float t = b - (c.x * c.x + c.y * c.y) * frcp(asafe);
  float i22 = frsq(fmaxf(t, 1e-20f));
  float2 I10 = cscalef(L10, -i11 * i22);
  // M = Linv * Gn ; C = M * Linv^H with Linv^H = [[i11, conj(I10)],[0, i22]]
  float2 M00 = make_float2(i11 * n00, 0.f);
  float2 M01 = cscalef(n01, i11);
  float2 M10 = caddf(cscalef(I10, n00), cscalef(conjf2(n01), i22));
  float2 M11 = caddf(cmulf(I10, n01), make_float2(i22 * n11, 0.f));
  float p = M00.x * i11;
  float2 q = caddf(cmulf(M00, conjf2(I10)), cscalef(M01, i22));
  float r = caddf(cmulf(M10, conjf2(I10)), cscalef(M11, i22)).x;
  float qq = q.x * q.x + q.y * q.y;
  float s = fsqrt(fmaxf((p - r) * (p - r) + 4.f * qq, 0.f));
  float lam0 = 0.5f * ((p + r) - s);
  float2 e0a, e0b;
  if (qq > 1e-30f) {
    e0a = q;
    e0b = make_float2(lam0 - p, 0.f);
  } else if (p <= r) {
    e0a = make_float2(1.f, 0.f); e0b = make_float2(0.f, 0.f);
  } else {
    e0a = make_float2(0.f, 0.f); e0b = make_float2(1.f, 0.f);
  }
  float inv_n = frcp(fmaxf(fsqrt(e0a.x * e0a.x + e0a.y * e0a.y + e0b.x * e0b.x + e0b.y * e0b.y), 1e-30f));
  e0a = cscalef(e0a, inv_n);
  e0b = cscalef(e0b, inv_n);
  float2 e1a = cscalef(conjf2(e0b), -1.f);
  float2 e1b = conjf2(e0a);
  // Vg columns: h = Linv^H * e
  float2 h0_0 = caddf(cscalef(e0a, i11), cmulf(conjf2(I10), e0b));
  float2 h0_1 = cscalef(e0b, i22);
  float2 h1_0 = caddf(cscalef(e1a, i11), cmulf(conjf2(I10), e1b));
  float2 h1_1 = cscalef(e1b, i22);
  float quad0 = a * (h0_0.x * h0_0.x + h0_0.y * h0_0.y) + b * (h0_1.x * h0_1.x + h0_1.y * h0_1.y) +
                2.f * cmulf(cmulf(conjf2(h0_0), c), h0_1).x;
  float quad1 = n00 * (h1_0.x * h1_0.x + h1_0.y * h1_0.y) + n11 * (h1_1.x * h1_1.x + h1_1.y * h1_1.y) +
                2.f * cmulf(cmulf(conjf2(h1_0), n01), h1_1).x;
  float s0 = frsq(fmaxf(quad0, EPS));
  float s1 = frsq(fmaxf(quad1, EPS));
  float* w = Wm + (size_t)i * 8;  // [W00 W01 W10 W11] as re,im pairs
  float2 W00 = cscalef(h0_0, s0), W10 = cscalef(h0_1, s0);
  float2 W01 = cscalef(h1_0, s1), W11 = cscalef(h1_1, s1);
  w[0] = W00.x; w[1] = W00.y; w[2] = W01.x; w[3] = W01.y;
  w[4] = W10.x; w[5] = W10.y; w[6] = W11.x; w[7] = W11.y;
}

// ---- K8: Y[i,m,j] = conj(W[0][m]) x0 + conj(W[1][m]) x1 -------------------
__global__ __launch_bounds__(256) void k_applyW(const float2* __restrict__ Xc,
                                                const float* __restrict__ Wm,
                                                float2* __restrict__ Y) {
  size_t t = (size_t)blockIdx.x * 256 + threadIdx.x;
  if (t >= (size_t)I_N * J_N) return;
  int i = (int)(t / J_N), j = (int)(t % J_N);
  const float* w = Wm + (size_t)i * 8;
  float2 w00 = make_float2(w[0], w[1]), w01 = make_float2(w[2], w[3]);
  float2 w10 = make_float2(w[4], w[5]), w11 = make_float2(w[6], w[7]);
  float2 x0 = Xc[((size_t)i * 2) * J_N + j];
  float2 x1 = Xc[((size_t)i * 2 + 1) * J_N + j];
  Y[((size_t)i * 2) * J_N + j] = caddf(conjmul(w00, x0), conjmul(w10, x1));
  Y[((size_t)i * 2 + 1) * J_N + j] = caddf(conjmul(w01, x0), conjmul(w11, x1));
}

// ---- K10/K11: elementwise r_s / r_n updates (it >= SWITCH_ITER) -----------
__global__ __launch_bounds__(256) void k_rs_update(float* __restrict__ rs,
                                                   const float* __restrict__ rn,
                                                   const float* __restrict__ p1,
                                                   const float* __restrict__ lam) {
  size_t t = (size_t)blockIdx.x * 256 + threadIdx.x;
  if (t >= (size_t)I_N * J_N) return;
  int i = (int)(t / J_N);
  float vrs = rs[t], vrn = rn[t];
  float il = frcp(lam_of(vrs, vrn, lam[i]));
  float irs = frcp(vrs + EPS);
  float num = fmaf(p1[t], il * il, BETA * irs * irs);
  float den = il + (ALPHA + 1.0f) * irs + EPS;
  rs[t] = fmaxf(vrs * fsqrt(num * frcp(den)), EPS);
}

__global__ __launch_bounds__(256) void k_rn_update(const float* __restrict__ rs,
                                                   float* __restrict__ rn,
                                                   const float* __restrict__ p1,
                                                   const float* __restrict__ pn,
                                                   const float* __restrict__ lam) {
  size_t t = (size_t)blockIdx.x * 256 + threadIdx.x;
  if (t >= (size_t)I_N * J_N) return;
  int i = (int)(t / J_N);
  float lv = lam[i];
  float vrs = rs[t], vrn = rn[t];
  float il = frcp(lam_of(vrs, vrn, lv));
  float irn = frcp(vrn + EPS);
  float num = fmaf(lv * p1[t], il * il, pn[t] * irn * irn);
  float den = fmaf(lv, il, irn) + EPS;  // (M-1)==1
  rn[t] = fmaxf(vrn * fsqrt(num * frcp(den)), EPS);
}

// ---- K9: output S broadcast to (M,J,I,2); coalesced writes over i ---------
__global__ __launch_bounds__(256) void k_finalize(const float2* __restrict__ Y,
                                                  const float* __restrict__ rs,
                                                  const float* __restrict__ rn,
                                                  const float* __restrict__ lam,
                                                  float2* __restrict__ out) {
  int i = blockIdx.x * 256 + threadIdx.x;
  if (i >= I_N) return;
  int j = blockIdx.y;
  size_t t = (size_t)i * J_N + j;
  float vrs = rs[t];
  float g = vrs * frcp(lam_of(vrs, rn[t], lam[i]));
  float2 y = Y[((size_t)i * 2) * J_N + j];
  float2 sv = make_float2(g * y.x, g * y.y);
  out[((size_t)0 * J_N + j) * I_N + i] = sv;
  out[((size_t)1 * J_N + j) * I_N + i] = sv;
}

// ---------------------------------------------------------------------------
extern "C" void kernel_launch(void* const* d_in, const int* in_sizes, int n_in,
                              void* d_out, int out_size, void* d_ws, size_t ws_size,
                              hipStream_t stream) {
  (void)in_sizes; (void)n_in; (void)out_size; (void)ws_size;
  const size_t NIJ = (size_t)I_N * J_N;
  float* ws = (float*)d_ws;
  float2* Xc = (float2*)ws;            // 2*NIJ float2
  float2* Yb = Xc + 2 * NIJ;           // 2*NIJ float2
  float* rs = ws + 8 * NIJ;
  float* rn = rs + NIJ;
  float* p1 = rn + NIJ;
  float* pn = p1 + NIJ;
  float* sm = pn + NIJ;
  float* Ts = sm;
  float* Tn = Ts + 8448;
  float* Vs = Tn + 8448;
  float* Vn = Vs + 32768;
  float* lam = Vn + 32768;
  float* Gs = lam + 1056;
  float* Gn = Gs + 4352;
  float* Wm = Gn + 4352;

  const float2* X = (const float2*)d_in[0];
  const float* Ts_in = (const float*)d_in[1];
  const float* Vs_in = (const float*)d_in[2];
  const float* Tn_in = (const float*)d_in[3];
  const float* Vn_in = (const float*)d_in[4];

  const int EG = (int)((NIJ + 255) / 256);  // 16400 blocks for elementwise

  k_transpose_init<<<dim3(J_N / 16, (I_N + 15) / 16, 2), dim3(16, 16), 0, stream>>>(X, Xc, Yb);
  k_small_init<<<128, 256, 0, stream>>>(Ts, Tn, Vs, Vn, lam, Ts_in, Vs_in, Tn_in, Vn_in);
  k_power<<<EG, 256, 0, stream>>>(Yb, p1, pn);

  for (int it = 0; it < N_ITER; ++it) {
    if (it < SWITCH_ITER) {
      k_recompute_r<<<EG, 256, 0, stream>>>(1, 1, Ts, Vs, Tn, Vn, rs, rn);
      // T_s update, refresh r_s
      k_rowreduce<0><<<I_N, 256, 0, stream>>>(Ts, Vs, rs, rn, p1, pn, lam);
      k_recompute_r<<<EG, 256, 0, stream>>>(1, 0, Ts, Vs, Tn, Vn, rs, rn);
      // V_s update (WMMA), refresh r_s
      k_vupdate_wmma<0><<<32, 256, 0, stream>>>(Vs, Ts, rs, rn, p1, pn, lam);
      k_recompute_r<<<EG, 256, 0, stream>>>(1, 0, Ts, Vs, Tn, Vn, rs, rn);
      // T_n update, refresh r_n
      k_rowreduce<1><<<I_N, 256, 0, stream>>>(Tn, Vn, rs, rn, p1, pn, lam);
      k_recompute_r<<<EG, 256, 0, stream>>>(0, 1, Ts, Vs, Tn, Vn, rs, rn);
      // V_n update (WMMA), refresh r_n
      k_vupdate_wmma<1><<<32, 256, 0, stream>>>(Vn, Tn, rs, rn, p1, pn, lam);
      k_recompute_r<<<EG, 256, 0, stream>>>(0, 1, Ts, Vs, Tn, Vn, rs, rn);
      // lam_n update
      k_rowreduce<2><<<I_N, 256, 0, stream>>>(Ts, Vs, rs, rn, p1, pn, lam);
      // spatial model: G_s,G_n -> eig -> W -> Y -> powers
      k_covariance<<<I_N, 256, 0, stream>>>(Xc, rs, rn, lam, Gs, Gn);
      k_eig2x2<<<(I_N + 255) / 256, 256, 0, stream>>>(Gs, Gn, Wm);
      k_applyW<<<EG, 256, 0, stream>>>(Xc, Wm, Yb);
      k_power<<<EG, 256, 0, stream>>>(Yb, p1, pn);
    } else {
      k_rs_update<<<EG, 256, 0, stream>>>(rs, rn, p1, lam);
      k_rn_update<<<EG, 256, 0, stream>>>(rs, rn, p1, pn, lam);
      k_rowreduce<2><<<I_N, 256, 0, stream>>>(Ts, Vs, rs, rn, p1, pn, lam);
    }
  }
  k_finalize<<<dim3((I_N + 255) / 256, J_N), 256, 0, stream>>>(Yb, rs, rn, lam, (float2*)d_out);
}